// SoftAttention_2388001816847
// MI455X (gfx1250) — compile-verified
//
#include <hip/hip_runtime.h>
#include <hip/hip_bf16.h>

typedef __attribute__((ext_vector_type(2))) float v2f;
typedef __attribute__((ext_vector_type(4))) float v4f;
typedef __attribute__((ext_vector_type(8))) float v8f;

// Fused masked cross-attention (flash-attention style) for one direction:
//   Out[i,:] = masked_softmax_j( mq[i]*mk[j] , Q[i]·K[j] ) @ K
// Launched twice with (Q,K,mq,mk) swapped to produce attended_a and attended_b.
//
// One wave (32 lanes) per (batch, 16-row Q tile). L=1024, D=256 compile-time.
template <int L, int D>
__global__ __launch_bounds__(32, 1) void fused_masked_attn(
    const float* __restrict__ Q,   // [B, L, D]
    const float* __restrict__ KV,  // [B, L, D]  (keys == values)
    const int*   __restrict__ mask_q,  // [B, L]
    const int*   __restrict__ mask_k,  // [B, L]
    float* __restrict__ Out)       // [B, L, D]
{
    constexpr int TILES_PER_SEQ = L / 16;

    __shared__ float qs[16][D];     // Q tile, 16 KB
    __shared__ float vs[16][D];     // K/V chunk, 16 KB
    __shared__ float ps[16][16];    // S tile -> P tile
    __shared__ float rowScale[16];
    __shared__ float rowSum[16];
    __shared__ int   mks[16];

    const int bid   = blockIdx.x;
    const int batch = bid / TILES_PER_SEQ;
    const int i0    = (bid % TILES_PER_SEQ) * 16;

    const int lane = threadIdx.x;       // 0..31
    const int h    = lane >> 4;         // half-wave: K-pair select in WMMA frags
    const int l16  = lane & 15;         // M / N index

    const float* qbase = Q  + (size_t)(batch * L) * D;
    const float* vbase = KV + (size_t)(batch * L) * D;

    // ---- stage Q tile into LDS (coalesced float4) ----
    for (int t = lane; t < 16 * (D / 4); t += 32) {
        const int row = t / (D / 4);
        const int c4  = t % (D / 4);
        *(v4f*)&qs[row][c4 * 4] =
            *(const v4f*)&qbase[(size_t)(i0 + row) * D + c4 * 4];
    }

    // per-row (owned by lanes 0..15) online-softmax state
    const int   mq   = mask_q[batch * L + i0 + l16];
    float Mreg = -3.0e38f;   // finite sentinel: exp(Mold - Mnew) stays well-defined
    float Sreg = 0.0f;

    v8f acc[16];             // O accumulator: 16 column tiles of 16x16 f32
    const v8f vzero = {};
#pragma unroll
    for (int t = 0; t < 16; ++t) acc[t] = vzero;

    __syncthreads();

    // ================== chunk loop over K/V rows ==================
    for (int j0 = 0; j0 < L; j0 += 16) {
        // ---- stage K/V chunk + column mask into LDS ----
        for (int t = lane; t < 16 * (D / 4); t += 32) {
            const int row = t / (D / 4);
            const int c4  = t % (D / 4);
            *(v4f*)&vs[row][c4 * 4] =
                *(const v4f*)&vbase[(size_t)(j0 + row) * D + c4 * 4];
        }
        if (lane < 16) mks[lane] = mask_k[batch * L + j0 + lane];

        // ---- prefetch NEXT chunk (hides global latency behind 128 WMMAs) ----
        // next chunk = 16 rows x 1 KB = 128 cachelines of 128 B -> 4/lane.
        if (j0 + 16 < L) {
#pragma unroll
            for (int t = 0; t < 4; ++t) {
                const int idx = lane + 32 * t;        // 0..127
                const int row = idx >> 3;             // 0..15
                const int seg = idx & 7;              // 128-B segment in row
                __builtin_prefetch(
                    &vbase[(size_t)(j0 + 16 + row) * D + seg * 32], 0, 0);
            }
        }
        __syncthreads();

        // ---- GEMM1: S(16x16) = Qtile · Kchunk^T over D=256 (64 WMMAs) ----
        v8f s_acc = vzero;
#pragma unroll
        for (int s = 0; s < D / 4; ++s) {
            const v2f af = *(const v2f*)&qs[l16][4 * s + 2 * h];
            const v2f bf = *(const v2f*)&vs[l16][4 * s + 2 * h];
            s_acc = __builtin_amdgcn_wmma_f32_16x16x4_f32(
                false, af, false, bf, (short)0, s_acc, false, false);
        }

        // spill S tile to LDS (C layout: row = r + 8*h, col = l16)
#pragma unroll
        for (int r = 0; r < 8; ++r) ps[r + 8 * h][l16] = s_acc[r];
        __syncthreads();

        // ---- online masked softmax, one row per lane (lanes 0..15) ----
        if (lane < 16) {
            const int r = lane;
            float z[16];
            float mx = Mreg;
#pragma unroll
            for (int j = 0; j < 16; ++j) {
                z[j] = ps[r][j];
                const bool valid = (mq != 0) && (mks[j] != 0);
                mx = (valid && z[j] > mx) ? z[j] : mx;
            }
            const float scale = __expf(Mreg - mx);  // ==1 when mx unchanged
            float sums = 0.0f;
#pragma unroll
            for (int j = 0; j < 16; ++j) {
                const bool valid = (mq != 0) && (mks[j] != 0);
                const float w = valid ? __expf(z[j] - mx) : 0.0f;
                ps[r][j] = w;
                sums += w;
            }
            rowScale[r] = scale;
            Sreg = Sreg * scale + sums;
            Mreg = mx;
        }
        __syncthreads();

        // ---- rescale O accumulators by exp(Mold - Mnew) per row ----
#pragma unroll
        for (int r = 0; r < 8; ++r) {
            const float scl = rowScale[r + 8 * h];
#pragma unroll
            for (int t = 0; t < 16; ++t) acc[t][r] *= scl;
        }

        // ---- GEMM2: O(16x256) += P(16x16) · Vchunk(16x256)  (64 WMMAs) ----
#pragma unroll
        for (int ss = 0; ss < 4; ++ss) {
            const v2f pf = *(const v2f*)&ps[l16][4 * ss + 2 * h];
#pragma unroll
            for (int t = 0; t < 16; ++t) {
                v2f bf;
                bf[0] = vs[4 * ss + 2 * h + 0][16 * t + l16];
                bf[1] = vs[4 * ss + 2 * h + 1][16 * t + l16];
                acc[t] = __builtin_amdgcn_wmma_f32_16x16x4_f32(
                    false, pf, false, bf, (short)0, acc[t], false, false);
            }
        }
        __syncthreads();  // before next chunk overwrites vs/ps
    }

    // ================== epilogue: normalize + store ==================
    if (lane < 16) rowSum[lane] = Sreg;
    __syncthreads();

    float inv[8];
#pragma unroll
    for (int r = 0; r < 8; ++r)
        inv[r] = 1.0f / (rowSum[r + 8 * h] + 1e-13f);

    float* obase = Out + (size_t)(batch * L) * D;
#pragma unroll
    for (int t = 0; t < 16; ++t) {
#pragma unroll
        for (int r = 0; r < 8; ++r) {
            obase[(size_t)(i0 + r + 8 * h) * D + 16 * t + l16] = acc[t][r] * inv[r];
        }
    }
}

extern "C" void kernel_launch(void* const* d_in, const int* in_sizes, int n_in,
                              void* d_out, int out_size, void* d_ws, size_t ws_size,
                              hipStream_t stream) {
    constexpr int L = 1024;
    constexpr int D = 256;

    const float* a      = (const float*)d_in[0];
    const float* b      = (const float*)d_in[1];
    const int*   mask_a = (const int*)d_in[2];
    const int*   mask_b = (const int*)d_in[3];

    const int B = in_sizes[0] / (L * D);  // 32

    float* out_a = (float*)d_out;                       // attended_a [B, L, D]
    float* out_b = out_a + (size_t)B * L * D;           // attended_b [B, L, D]

    const int blocks = B * (L / 16);
    // Direction 1: rows of a attend over b.
    fused_masked_attn<L, D><<<blocks, 32, 0, stream>>>(a, b, mask_a, mask_b, out_a);
    // Direction 2 is the same computation with roles swapped.
    fused_masked_attn<L, D><<<blocks, 32, 0, stream>>>(b, a, mask_b, mask_a, out_b);
}